// DrugResponseModel_78065325572502
// MI455X (gfx1250) — compile-verified
//
#include <hip/hip_runtime.h>
#include <math.h>

// ---------------------------------------------------------------------------
// DrugResponseModel on gfx1250 (MI455X): 2x GATConv + mean-pool + MLPs.
// GEMMs use V_WMMA_F32_16X16X4_F32 (exact fp32, CDNA5 matrix pipe).
// Edge softmax/aggregation use global float atomics (memory-bound phase).
// ---------------------------------------------------------------------------

typedef float v2f __attribute__((ext_vector_type(2)));
typedef float v8f __attribute__((ext_vector_type(8)));

#define NEG_SLOPE 0.2f
#define BN_EPS    1e-5f

#define EPI_NONE   0
#define EPI_RELU   1
#define EPI_BNRELU 2

#define NN  65536
#define NE  262144
#define TOT (NE + NN)   // edges + self loops
#define NB  2048

__device__ __forceinline__ v8f wmma_f32_4(v2f a, v2f b, v8f c) {
  // 8 args: (neg_a, A, neg_b, B, c_mod, C, reuse_a, reuse_b)
  return __builtin_amdgcn_wmma_f32_16x16x4_f32(false, a, false, b, (short)0, c,
                                               false, false);
}

// ---------------------------------------------------------------------------
// C[M,N] = epilogue(A[M,K] @ B[K,N] + bias)   (all row-major, C has ldc)
// Block = 128 threads = 4 waves; wave owns a 32x64 tile; block = 128x64.
// Requires M % 128 == 0 and N % 64 == 0 (true for all call sites).
// Main K loop has NO bounds checks (unconditional loads); the K%4 tail is
// handled branchlessly with clamped addresses + 0/1 masks, so EXEC stays
// all-ones throughout (WMMA requirement) and no exec-predicated loads occur.
// ---------------------------------------------------------------------------
__global__ __launch_bounds__(128)
void gemm_wmma_f32(const float* __restrict__ A, const float* __restrict__ B,
                   const float* __restrict__ bias, float* __restrict__ C,
                   int M, int K, int N, int ldc, int epi,
                   const float* __restrict__ bn_gamma,
                   const float* __restrict__ bn_beta,
                   const float* __restrict__ bn_mean,
                   const float* __restrict__ bn_var) {
  const int lane = threadIdx.x & 31;
  const int wave = threadIdx.x >> 5;
  const int col0 = blockIdx.x * 64;
  const int row0 = blockIdx.y * 128 + wave * 32;
  const int mr   = lane & 15;            // M (for A) / N (for B) within tile
  const int koff = (lane >> 4) * 2;      // K sub-offset per half-wave

  v8f acc[2][4];
#pragma unroll
  for (int i = 0; i < 2; ++i)
#pragma unroll
    for (int s = 0; s < 4; ++s) {
      v8f z = {0.f, 0.f, 0.f, 0.f, 0.f, 0.f, 0.f, 0.f};
      acc[i][s] = z;
    }

  // Hoisted per-lane base pointers.
  const float* Ap0 = A + (size_t)(row0 + mr) * K + koff;
  const float* Ap1 = A + (size_t)(row0 + 16 + mr) * K + koff;
  const float* Bp  = B + (size_t)koff * N + (col0 + mr);

  const int Kmain = K & ~3;
  for (int k = 0; k < Kmain; k += 4) {
    v2f a0, a1;
    a0.x = Ap0[k];
    a0.y = Ap0[k + 1];
    a1.x = Ap1[k];
    a1.y = Ap1[k + 1];
    const float* Bk = Bp + (size_t)k * N;
#pragma unroll
    for (int s = 0; s < 4; ++s) {
      v2f b;
      b.x = Bk[s * 16];
      b.y = Bk[(size_t)N + s * 16];
      acc[0][s] = wmma_f32_4(a0, b, acc[0][s]);
      acc[1][s] = wmma_f32_4(a1, b, acc[1][s]);
    }
  }

  if (Kmain < K) {  // branchless tail for K % 4 != 0 (uniform branch)
    const int kb0 = Kmain + koff;
    const int kb1 = kb0 + 1;
    const float m0 = (kb0 < K) ? 1.f : 0.f;   // v_cndmask, not a branch
    const float m1 = (kb1 < K) ? 1.f : 0.f;
    const int   c0 = (kb0 < K) ? kb0 : 0;     // clamped (always-valid) addr
    const int   c1 = (kb1 < K) ? kb1 : 0;
    const size_t rA0 = (size_t)(row0 + mr) * K;
    const size_t rA1 = (size_t)(row0 + 16 + mr) * K;
    v2f a0, a1;
    a0.x = A[rA0 + c0] * m0;
    a0.y = A[rA0 + c1] * m1;
    a1.x = A[rA1 + c0] * m0;
    a1.y = A[rA1 + c1] * m1;
#pragma unroll
    for (int s = 0; s < 4; ++s) {
      const int n = col0 + s * 16 + mr;
      v2f b;
      b.x = B[(size_t)c0 * N + n] * m0;
      b.y = B[(size_t)c1 * N + n] * m1;
      acc[0][s] = wmma_f32_4(a0, b, acc[0][s]);
      acc[1][s] = wmma_f32_4(a1, b, acc[1][s]);
    }
  }

  // Epilogue. C/D layout: vgpr r, lanes 0-15 -> M=r, lanes 16-31 -> M=8+r.
#pragma unroll
  for (int i = 0; i < 2; ++i) {
    const int mbase = row0 + 16 * i + ((lane >> 4) << 3);
#pragma unroll
    for (int s = 0; s < 4; ++s) {
      const int n = col0 + s * 16 + mr;
      const float bv = bias ? bias[n] : 0.f;
      float bns = 1.f, bnb = 0.f, bnm = 0.f;
      if (epi == EPI_BNRELU) {
        bns = bn_gamma[n] * rsqrtf(bn_var[n] + BN_EPS);
        bnb = bn_beta[n];
        bnm = bn_mean[n];
      }
#pragma unroll
      for (int r = 0; r < 8; ++r) {
        float v = acc[i][s][r] + bv;
        if (epi == EPI_BNRELU) v = (v - bnm) * bns + bnb;
        if (epi != EPI_NONE) v = fmaxf(v, 0.f);
        C[(size_t)(mbase + r) * ldc + n] = v;
      }
    }
  }
}

// ---------------------------------------------------------------------------
// Per-node attention logits: a_src[n,h] = <feat[n, h*64:(h+1)*64], att_src[h]>
// One wave per node; 4 lanes per head (wave32).
// ---------------------------------------------------------------------------
__global__ __launch_bounds__(256)
void att_scores(const float* __restrict__ feat, const float* __restrict__ att_s,
                const float* __restrict__ att_d, float* __restrict__ a_src,
                float* __restrict__ a_dst, int nnodes) {
  const int gw   = (blockIdx.x * blockDim.x + threadIdx.x) >> 5;
  const int lane = threadIdx.x & 31;
  if (gw >= nnodes) return;
  const float* hr = feat + (size_t)gw * 512 + lane * 16;
  const float* sp = att_s + lane * 16;
  const float* dp = att_d + lane * 16;
  float ps = 0.f, pd = 0.f;
#pragma unroll
  for (int i = 0; i < 16; i += 4) {
    float4 hv = *(const float4*)(hr + i);
    float4 sv = *(const float4*)(sp + i);
    float4 dv = *(const float4*)(dp + i);
    ps += hv.x * sv.x + hv.y * sv.y + hv.z * sv.z + hv.w * sv.w;
    pd += hv.x * dv.x + hv.y * dv.y + hv.z * dv.z + hv.w * dv.w;
  }
  ps += __shfl_xor(ps, 1); ps += __shfl_xor(ps, 2);
  pd += __shfl_xor(pd, 1); pd += __shfl_xor(pd, 2);
  if ((lane & 3) == 0) {
    const int head = lane >> 2;
    a_src[(size_t)gw * 8 + head] = ps;
    a_dst[(size_t)gw * 8 + head] = pd;
  }
}

// Order-independent float atomic max via signed/unsigned integer ordering.
// Init pattern 0xFFFFFFFF loses to every real value under this ordering.
__device__ __forceinline__ void atomicMaxF(float* addr, float v) {
  if (v >= 0.f) atomicMax((int*)addr, __float_as_int(v));
  else          atomicMin((unsigned int*)addr, __float_as_uint(v));
}

__global__ __launch_bounds__(256)
void edge_lrelu_max(const int* __restrict__ ei, const float* __restrict__ a_src,
                    const float* __restrict__ a_dst, float* __restrict__ ev,
                    float* __restrict__ emax) {
  const int idx = blockIdx.x * blockDim.x + threadIdx.x;
  if (idx >= TOT * 8) return;
  const int e  = idx >> 3;
  const int hd = idx & 7;
  const int src = (e < NE) ? ei[e]      : (e - NE);
  const int dst = (e < NE) ? ei[NE + e] : (e - NE);
  float v = a_src[(size_t)src * 8 + hd] + a_dst[(size_t)dst * 8 + hd];
  v = (v > 0.f) ? v : NEG_SLOPE * v;
  ev[idx] = v;
  atomicMaxF(&emax[(size_t)dst * 8 + hd], v);
}

__global__ __launch_bounds__(256)
void edge_exp_sum(const int* __restrict__ ei, const float* __restrict__ ev,
                  const float* __restrict__ emax, float* __restrict__ alpha,
                  float* __restrict__ denom) {
  const int idx = blockIdx.x * blockDim.x + threadIdx.x;
  if (idx >= TOT * 8) return;
  const int e  = idx >> 3;
  const int hd = idx & 7;
  const int dst = (e < NE) ? ei[NE + e] : (e - NE);
  const float ex = expf(ev[idx] - emax[(size_t)dst * 8 + hd]);
  alpha[idx] = ex;
  atomicAdd(&denom[(size_t)dst * 8 + hd], ex);
}

// One block per edge; 256 threads cover 512 channels (2 each, float2 loads).
__global__ __launch_bounds__(256)
void edge_message(const int* __restrict__ ei, const float* __restrict__ hsrc,
                  const float* __restrict__ alpha, const float* __restrict__ denom,
                  float* __restrict__ out) {
  const int e = blockIdx.x;
  const int t = threadIdx.x;
  const int src = (e < NE) ? ei[e]      : (e - NE);
  const int dst = (e < NE) ? ei[NE + e] : (e - NE);
  const int head = t >> 5;                        // (t*2)/64
  const float al = alpha[(size_t)e * 8 + head] /
                   (denom[(size_t)dst * 8 + head] + 1e-16f);
  const int c = t * 2;
  const float2 hv = *(const float2*)(hsrc + (size_t)src * 512 + c);
  atomicAdd(out + (size_t)dst * 512 + c,     hv.x * al);
  atomicAdd(out + (size_t)dst * 512 + c + 1, hv.y * al);
}

__global__ __launch_bounds__(256)
void bias_elu(const float* __restrict__ in, const float* __restrict__ bias,
              float* __restrict__ out, size_t n) {
  const size_t i = (size_t)blockIdx.x * blockDim.x + threadIdx.x;
  if (i >= n) return;
  const float v = in[i] + bias[i & 511];
  out[i] = (v > 0.f) ? v : (expf(v) - 1.f);
}

__global__ __launch_bounds__(256)
void pool_count(const int* __restrict__ batch, float* __restrict__ cnt) {
  const int i = blockIdx.x * blockDim.x + threadIdx.x;
  if (i >= NN) return;
  atomicAdd(&cnt[batch[i]], 1.0f);
}

__global__ __launch_bounds__(256)
void pool_sum(const float* __restrict__ feat, const int* __restrict__ batch,
              float* __restrict__ csum /* ld = 1024 */) {
  const size_t i = (size_t)blockIdx.x * blockDim.x + threadIdx.x;
  if (i >= (size_t)NN * 512) return;
  const int n = (int)(i >> 9);
  const int c = (int)(i & 511);
  atomicAdd(&csum[(size_t)batch[n] * 1024 + c], feat[i]);
}

__global__ __launch_bounds__(256)
void pool_div(float* __restrict__ cbuf /* ld = 1024 */,
              const float* __restrict__ cnt) {
  const int i = blockIdx.x * blockDim.x + threadIdx.x;
  if (i >= NB * 512) return;
  const int g = i >> 9;
  const int c = i & 511;
  cbuf[(size_t)g * 1024 + c] /= fmaxf(cnt[g], 1.0f);
}

// Final Linear(128 -> 1): one wave per row.
__global__ __launch_bounds__(256)
void final_dot(const float* __restrict__ f2, const float* __restrict__ w,
               const float* __restrict__ b, float* __restrict__ out, int M) {
  const int gw   = (blockIdx.x * blockDim.x + threadIdx.x) >> 5;
  const int lane = threadIdx.x & 31;
  if (gw >= M) return;
  const float4 x  = *(const float4*)(f2 + (size_t)gw * 128 + lane * 4);
  const float4 wv = *(const float4*)(w + lane * 4);
  float p = x.x * wv.x + x.y * wv.y + x.z * wv.z + x.w * wv.w;
#pragma unroll
  for (int o = 16; o > 0; o >>= 1) p += __shfl_xor(p, o);
  if (lane == 0) out[gw] = p + b[0];
}

// ---------------------------------------------------------------------------
extern "C" void kernel_launch(void* const* d_in, const int* in_sizes, int n_in,
                              void* d_out, int out_size, void* d_ws, size_t ws_size,
                              hipStream_t stream) {
  (void)in_sizes; (void)n_in; (void)out_size; (void)ws_size;
  const float* x     = (const float*)d_in[0];
  const int*   ei    = (const int*)  d_in[1];
  const int*   batch = (const int*)  d_in[2];
  const float* x_gen = (const float*)d_in[3];
  const float* W1    = (const float*)d_in[4];
  const float* attS1 = (const float*)d_in[5];
  const float* attD1 = (const float*)d_in[6];
  const float* bias1 = (const float*)d_in[7];
  const float* W2    = (const float*)d_in[8];
  const float* attS2 = (const float*)d_in[9];
  const float* attD2 = (const float*)d_in[10];
  const float* bias2 = (const float*)d_in[11];
  const float* gW1   = (const float*)d_in[12];
  const float* gb1   = (const float*)d_in[13];
  const float* bn_g  = (const float*)d_in[14];
  const float* bn_b  = (const float*)d_in[15];
  const float* bn_m  = (const float*)d_in[16];
  const float* bn_v  = (const float*)d_in[17];
  const float* gW2   = (const float*)d_in[18];
  const float* gb2   = (const float*)d_in[19];
  const float* fW1   = (const float*)d_in[20];
  const float* fb1   = (const float*)d_in[21];
  const float* fW2   = (const float*)d_in[22];
  const float* fb2   = (const float*)d_in[23];
  const float* fW3   = (const float*)d_in[24];
  const float* fb3   = (const float*)d_in[25];

  char* p = (char*)d_ws;
  auto alloc = [&](size_t bytes) -> char* {
    char* r = p;
    p += (bytes + 255) & ~(size_t)255;
    return r;
  };
  float* hbuf  = (float*)alloc((size_t)NN * 512 * 4);   // node feats ping
  float* gbuf  = (float*)alloc((size_t)NN * 512 * 4);   // node feats pong
  float* asrc  = (float*)alloc((size_t)NN * 8 * 4);
  float* adst  = (float*)alloc((size_t)NN * 8 * 4);
  float* emax  = (float*)alloc((size_t)NN * 8 * 4);
  float* denom = (float*)alloc((size_t)NN * 8 * 4);
  float* evb   = (float*)alloc((size_t)TOT * 8 * 4);
  float* alpb  = (float*)alloc((size_t)TOT * 8 * 4);
  float* cbuf  = (float*)alloc((size_t)NB * 1024 * 4);  // concat [g | z]
  float* z1    = (float*)alloc((size_t)NB * 128 * 4);
  float* f1    = (float*)alloc((size_t)NB * 256 * 4);
  float* f2    = (float*)alloc((size_t)NB * 128 * 4);
  float* cnt   = (float*)alloc((size_t)NB * 4);

  auto gemm = [&](const float* A, const float* Bm, const float* bias, float* C,
                  int M, int K, int N, int ldc, int epi,
                  const float* g = nullptr, const float* be = nullptr,
                  const float* mn = nullptr, const float* vr = nullptr) {
    dim3 grid((unsigned)(N / 64), (unsigned)(M / 128));
    hipLaunchKernelGGL(gemm_wmma_f32, grid, dim3(128), 0, stream,
                       A, Bm, bias, C, M, K, N, ldc, epi, g, be, mn, vr);
  };

  const int tot8_blocks = (TOT * 8 + 255) / 256;
  const size_t nfeat = (size_t)NN * 512;
  const int nfeat_blocks = (int)((nfeat + 255) / 256);

  // ---------------- GAT layer 1 ----------------
  gemm(x, W1, nullptr, hbuf, NN, 78, 512, 512, EPI_NONE);
  att_scores<<<NN / 8, 256, 0, stream>>>(hbuf, attS1, attD1, asrc, adst, NN);
  hipMemsetAsync(emax, 0xFF, (size_t)NN * 8 * 4, stream);
  hipMemsetAsync(denom, 0, (size_t)NN * 8 * 4, stream);
  hipMemsetAsync(gbuf, 0, nfeat * 4, stream);
  edge_lrelu_max<<<tot8_blocks, 256, 0, stream>>>(ei, asrc, adst, evb, emax);
  edge_exp_sum<<<tot8_blocks, 256, 0, stream>>>(ei, evb, emax, alpb, denom);
  edge_message<<<TOT, 256, 0, stream>>>(ei, hbuf, alpb, denom, gbuf);
  bias_elu<<<nfeat_blocks, 256, 0, stream>>>(gbuf, bias1, hbuf, nfeat);

  // ---------------- GAT layer 2 ----------------
  gemm(hbuf, W2, nullptr, gbuf, NN, 512, 512, 512, EPI_NONE);
  att_scores<<<NN / 8, 256, 0, stream>>>(gbuf, attS2, attD2, asrc, adst, NN);
  hipMemsetAsync(emax, 0xFF, (size_t)NN * 8 * 4, stream);
  hipMemsetAsync(denom, 0, (size_t)NN * 8 * 4, stream);
  hipMemsetAsync(hbuf, 0, nfeat * 4, stream);
  edge_lrelu_max<<<tot8_blocks, 256, 0, stream>>>(ei, asrc, adst, evb, emax);
  edge_exp_sum<<<tot8_blocks, 256, 0, stream>>>(ei, evb, emax, alpb, denom);
  edge_message<<<TOT, 256, 0, stream>>>(ei, gbuf, alpb, denom, hbuf);
  bias_elu<<<nfeat_blocks, 256, 0, stream>>>(hbuf, bias2, gbuf, nfeat);

  // ---------------- global mean pool -> cbuf[:, 0:512] ----------------
  hipMemsetAsync(cbuf, 0, (size_t)NB * 1024 * 4, stream);
  hipMemsetAsync(cnt, 0, (size_t)NB * 4, stream);
  pool_count<<<NN / 256, 256, 0, stream>>>(batch, cnt);
  pool_sum<<<nfeat_blocks, 256, 0, stream>>>(gbuf, batch, cbuf);
  pool_div<<<(NB * 512) / 256, 256, 0, stream>>>(cbuf, cnt);

  // ---------------- genomic encoder -> cbuf[:, 512:1024] ----------------
  gemm(x_gen, gW1, gb1, z1, NB, 735, 128, 128, EPI_BNRELU, bn_g, bn_b, bn_m, bn_v);
  gemm(z1, gW2, gb2, cbuf + 512, NB, 128, 512, 1024, EPI_RELU);

  // ---------------- fusion MLP ----------------
  gemm(cbuf, fW1, fb1, f1, NB, 1024, 256, 256, EPI_RELU);
  gemm(f1, fW2, fb2, f2, NB, 256, 128, 128, EPI_RELU);
  final_dot<<<NB / 8, 256, 0, stream>>>(f2, fW3, fb3, (float*)d_out, NB);
}